// UnsupervisedDiffusionLoss_62792421867865
// MI455X (gfx1250) — compile-verified
//
#include <hip/hip_runtime.h>
#include <stdint.h>
#include <math.h>

// Problem constants (from reference): B=4, N=4096, C_CONT=512, C_STYLE=256
constexpr int BB  = 4;
constexpr int NN  = 4096;
constexpr int TPB = 256;

typedef __attribute__((ext_vector_type(2))) float v2f;
typedef __attribute__((ext_vector_type(8))) float v8f;

// ---------------------------------------------------------------------------
// Deterministic block tree-reduction (no atomics -> bitwise stable results).
// ---------------------------------------------------------------------------
__device__ __forceinline__ float blockReduceSum(float v, float* red) {
    const int tid = threadIdx.x;
    __syncthreads();              // protect shared buffer reuse
    red[tid] = v;
    __syncthreads();
    #pragma unroll
    for (int s = TPB / 2; s > 0; s >>= 1) {
        if (tid < s) red[tid] += red[tid + s];
        __syncthreads();
    }
    return red[0];
}

// ---------------------------------------------------------------------------
// Chamfer (L1) kernel. grid = (NN/TPB, BB, 2). blockIdx.z picks direction:
//   z=0: query=generated, db=original  -> pA
//   z=1: query=original,  db=generated -> pB
// The entire 4096-point db set for the batch (48KB) is staged into LDS with
// CDNA5 async global->LDS copies, then each thread scans it (broadcast reads).
// ---------------------------------------------------------------------------
__global__ void chamfer_kernel(const float* __restrict__ gen,
                               const float* __restrict__ orig,
                               float* __restrict__ pA,
                               float* __restrict__ pB) {
    __shared__ __align__(16) float ldsdb[NN * 3];   // 48 KB
    __shared__ float red[TPB];

    const int b = blockIdx.y;
    const int z = blockIdx.z;
    const float* q    = z ? orig : gen;
    const float* dsrc = z ? gen  : orig;
    const float* dbg  = dsrc + (size_t)b * (NN * 3);

    // --- async stage: 49152 bytes = 12 iters * 256 threads * 16B ---
    unsigned ldsBase = (unsigned)(uintptr_t)&ldsdb[0];
    uint64_t gbase   = (uint64_t)(uintptr_t)dbg;
    #pragma unroll
    for (int it = 0; it < (NN * 3 * 4) / (TPB * 16); ++it) {
        unsigned off = (unsigned)((it * TPB + threadIdx.x) * 16);
        asm volatile("global_load_async_to_lds_b128 %0, %1, %2 offset:0"
                     :: "v"(ldsBase + off), "v"(off), "s"(gbase)
                     : "memory");
    }
    asm volatile("s_wait_asynccnt 0" ::: "memory");
    __syncthreads();

    // --- each thread: one query point, min over all 4096 db points ---
    const int i = blockIdx.x * TPB + threadIdx.x;
    const float* qp = q + ((size_t)b * NN + i) * 3;
    const float gx = qp[0], gy = qp[1], gz = qp[2];
    float m = 3.4e38f;
    #pragma unroll 4
    for (int j = 0; j < NN; ++j) {
        float dx = fabsf(gx - ldsdb[3 * j + 0]);
        float dy = fabsf(gy - ldsdb[3 * j + 1]);
        float dz = fabsf(gz - ldsdb[3 * j + 2]);
        m = fminf(m, dx + dy + dz);
    }

    float s = blockReduceSum(m, red);
    if (threadIdx.x == 0) {
        float* outp = z ? pB : pA;
        outp[b * gridDim.x + blockIdx.x] = s;
    }
}

// ---------------------------------------------------------------------------
// Generic SSE (sum of squared errors) partial reduction. out[blockIdx.x].
// ---------------------------------------------------------------------------
__global__ void mse_partial_kernel(const float* __restrict__ a,
                                   const float* __restrict__ bsrc,
                                   int n, float* __restrict__ out) {
    __shared__ float red[TPB];
    const int stride = gridDim.x * TPB;
    float s = 0.f;
    for (int i = blockIdx.x * TPB + threadIdx.x; i < n; i += stride) {
        float d = a[i] - bsrc[i];
        s += d * d;
    }
    float t = blockReduceSum(s, red);
    if (threadIdx.x == 0) out[blockIdx.x] = t;
}

// ---------------------------------------------------------------------------
// z-coordinate statistics (sum, sum of squares) for generated & original.
// grid = 64 x 256 threads == BB*NN points exactly.
// ---------------------------------------------------------------------------
__global__ void zstats_kernel(const float* __restrict__ g,
                              const float* __restrict__ o,
                              float* __restrict__ pGz,  float* __restrict__ pGz2,
                              float* __restrict__ pOz,  float* __restrict__ pOz2) {
    __shared__ float red[TPB];
    const int idx = blockIdx.x * TPB + threadIdx.x;   // 0..16383
    const float gz = g[3 * idx + 2];
    const float oz = o[3 * idx + 2];
    float t;
    t = blockReduceSum(gz, red);       if (threadIdx.x == 0) pGz [blockIdx.x] = t;
    t = blockReduceSum(gz * gz, red);  if (threadIdx.x == 0) pGz2[blockIdx.x] = t;
    t = blockReduceSum(oz, red);       if (threadIdx.x == 0) pOz [blockIdx.x] = t;
    t = blockReduceSum(oz * oz, red);  if (threadIdx.x == 0) pOz2[blockIdx.x] = t;
}

// ---------------------------------------------------------------------------
// Smoothness: per batch, sum & sumsq of ||g[i+1]-g[i]||_2 over 4095 segs.
// ---------------------------------------------------------------------------
__global__ void smooth_kernel(const float* __restrict__ g,
                              float* __restrict__ pS, float* __restrict__ pS2) {
    __shared__ float red[TPB];
    const int b = blockIdx.x;
    const float* base = g + (size_t)b * NN * 3;
    float s = 0.f, s2 = 0.f;
    for (int i = threadIdx.x; i < NN - 1; i += TPB) {
        float dx = base[3 * (i + 1) + 0] - base[3 * i + 0];
        float dy = base[3 * (i + 1) + 1] - base[3 * i + 1];
        float dz = base[3 * (i + 1) + 2] - base[3 * i + 2];
        float dn = sqrtf(dx * dx + dy * dy + dz * dz);
        s += dn; s2 += dn * dn;
    }
    float t = blockReduceSum(s, red);   if (threadIdx.x == 0) pS [b] = t;
    t       = blockReduceSum(s2, red);  if (threadIdx.x == 0) pS2[b] = t;
}

// ---------------------------------------------------------------------------
// WMMA-based 64-element sum: arrange 64 floats as a 16x4 A tile, multiply by
// all-ones B (4x16) -> D rows hold row-sums replicated across all 16 columns.
// Sum every lane/VGPR and divide by 16 (exact) => total. One wave (32 lanes).
// ---------------------------------------------------------------------------
__device__ __forceinline__ float wmma_sum64(const float* __restrict__ p) {
    const int lane = threadIdx.x;         // 0..31, EXEC all ones
    const int m  = lane & 15;
    const int k0 = (lane >> 4) << 1;      // lanes 0-15: K=0,1 ; lanes 16-31: K=2,3
    v2f a;
    a.x = p[m * 4 + k0];
    a.y = p[m * 4 + k0 + 1];
    v2f bones; bones.x = 1.0f; bones.y = 1.0f;
    v8f c = {0.f, 0.f, 0.f, 0.f, 0.f, 0.f, 0.f, 0.f};
    c = __builtin_amdgcn_wmma_f32_16x16x4_f32(
            /*neg_a=*/false, a, /*neg_b=*/false, bones,
            /*c_mod=*/(short)0, c, /*reuse_a=*/false, /*reuse_b=*/false);
    float s = c[0] + c[1] + c[2] + c[3] + c[4] + c[5] + c[6] + c[7];
    #pragma unroll
    for (int off = 16; off > 0; off >>= 1)
        s += __shfl_xor(s, off, 32);
    return s * 0.0625f;                   // each row-sum replicated 16x
}

// ---------------------------------------------------------------------------
// Finalize: combine all partials (one wave; scalar math in f64 on lane 0).
// Workspace layout (floats):
//   [0,64)   pA chamfer gen->orig      [64,128)  pB chamfer orig->gen
//   [128,192) pDiff SSE                [192,256) pGz   [256,320) pGz2
//   [320,384) pOz   [384,448) pOz2     [448] SSE cont  [449] SSE style
//   [456,460) smooth sums per batch    [460,464) smooth sumsqs per batch
// ---------------------------------------------------------------------------
__global__ void finalize_kernel(const float* __restrict__ ws,
                                const float* __restrict__ warm,
                                float* __restrict__ out) {
    float sumA = wmma_sum64(ws + 0);
    float sumB = wmma_sum64(ws + 64);
    float sse  = wmma_sum64(ws + 128);
    float sGz  = wmma_sum64(ws + 192);
    float sGz2 = wmma_sum64(ws + 256);
    float sOz  = wmma_sum64(ws + 320);
    float sOz2 = wmma_sum64(ws + 384);

    if (threadIdx.x == 0) {
        const double nz = (double)(BB * NN);
        double l_diff = (double)sse / (double)(BB * NN * 3);
        double l_cham = ((double)sumA + (double)sumB) / (2.0 * NN * BB);
        double l_cont = (double)ws[448] / (double)(BB * 512);
        double l_style = (double)ws[449] / (double)(BB * 256);

        double mg = (double)sGz / nz, mo = (double)sOz / nz;
        double vg = ((double)sGz2 - (double)sGz * (double)sGz / nz) / (nz - 1.0);
        double vo = ((double)sOz2 - (double)sOz * (double)sOz / nz) / (nz - 1.0);
        if (vg < 0.0) vg = 0.0;
        if (vo < 0.0) vo = 0.0;
        double l_lidar = fabs(mg - mo) + fabs(sqrt(vg) - sqrt(vo));

        const double nd = (double)(NN - 1);
        double ls = 0.0;
        for (int b = 0; b < BB; ++b) {
            double s  = (double)ws[456 + b];
            double s2 = (double)ws[460 + b];
            double var = (s2 - s * s / nd) / (nd - 1.0);
            if (var < 0.0) var = 0.0;
            ls += sqrt(var);
        }
        double l_smooth = ls / (double)BB;

        double w = (double)warm[0];
        double total = 1.0 * l_diff + w * (5.0 * l_cham + 2.0 * l_cont +
                                           0.5 * l_lidar + 0.01 * l_style +
                                           0.1 * l_smooth);
        out[0] = (float)l_diff;
        out[1] = (float)l_cham;
        out[2] = (float)l_cont;
        out[3] = (float)l_lidar;
        out[4] = (float)l_style;
        out[5] = (float)l_smooth;
        out[6] = (float)total;
    }
}

// ---------------------------------------------------------------------------
extern "C" void kernel_launch(void* const* d_in, const int* in_sizes, int n_in,
                              void* d_out, int out_size, void* d_ws, size_t ws_size,
                              hipStream_t stream) {
    const float* pred  = (const float*)d_in[0];
    const float* targ  = (const float*)d_in[1];
    const float* gen   = (const float*)d_in[2];
    const float* orig  = (const float*)d_in[3];
    const float* contO = (const float*)d_in[4];
    const float* contN = (const float*)d_in[5];
    const float* styS  = (const float*)d_in[6];
    const float* styT  = (const float*)d_in[7];
    const float* warm  = (const float*)d_in[8];
    float* ws  = (float*)d_ws;    // needs < 2 KB; every slot read is written each call
    float* out = (float*)d_out;

    chamfer_kernel<<<dim3(NN / TPB / 16 * 16 ? 16 : 16, BB, 2), TPB, 0, stream>>>(
        gen, orig, ws + 0, ws + 64);
    mse_partial_kernel<<<64, TPB, 0, stream>>>(pred, targ, BB * NN * 3, ws + 128);
    zstats_kernel<<<(BB * NN) / TPB, TPB, 0, stream>>>(gen, orig,
        ws + 192, ws + 256, ws + 320, ws + 384);
    mse_partial_kernel<<<1, TPB, 0, stream>>>(contN, contO, BB * 512, ws + 448);
    mse_partial_kernel<<<1, TPB, 0, stream>>>(styS, styT, BB * 256, ws + 449);
    smooth_kernel<<<BB, TPB, 0, stream>>>(gen, ws + 456, ws + 460);
    finalize_kernel<<<1, 32, 0, stream>>>(ws, warm, out);
}